// MultiheadAttention_24464133718580
// MI455X (gfx1250) — compile-verified
//
#include <hip/hip_runtime.h>
#include <hip/hip_bf16.h>

typedef __attribute__((ext_vector_type(2))) float v2f;
typedef __attribute__((ext_vector_type(8))) float v8f;
typedef __attribute__((ext_vector_type(4))) float f4;

#define B_  64
#define T_  4096
#define DK  128
#define D_  256
#define H_  8

// ---- workspace layout (float offsets) ----
// wv    : [B][16][128]   folded per-head score vectors (heads 8..15 zero pad)
// cvec  : [B][8]         folded bias dot products
// rmax  : [B][8]         softmax row max
// rsum  : [B][8]         1 / softmax denominator
// upart : [B][8 chunks][8][256]  partial u = sum_t attn * v
// sbuf  : [B][8][T]      masked scores
static constexpr size_t WV_OFF    = 0;
static constexpr size_t C_OFF     = WV_OFF    + (size_t)B_ * 16 * 128;   // 131072
static constexpr size_t RMAX_OFF  = C_OFF     + (size_t)B_ * 8;          // +512
static constexpr size_t RSUM_OFF  = RMAX_OFF  + (size_t)B_ * 8;          // +512
static constexpr size_t UPART_OFF = RSUM_OFF  + (size_t)B_ * 8;          // +512
static constexpr size_t S_OFF     = UPART_OFF + (size_t)B_ * 8 * 8 * 256;// +1048576
// total: S_OFF + B*8*T = ~3.28M floats (~13.1 MB)

// ---------------- K0: per-batch prep ----------------
__global__ void __launch_bounds__(128)
prep_kernel(const float* __restrict__ q, const float* __restrict__ Wq,
            const float* __restrict__ bq, const float* __restrict__ Wk,
            const float* __restrict__ bk, float* __restrict__ ws) {
  int b = blockIdx.x, tid = threadIdx.x;
  __shared__ float qh[DK];
  const float* qb = q + (size_t)b * DK;
  const float* wr = Wq + (size_t)tid * DK;
  float acc = bq[tid];
  #pragma unroll 4
  for (int j = 0; j < DK; ++j) acc = fmaf(wr[j], qb[j], acc);
  qh[tid] = acc;
  __syncthreads();
  // folded score vectors: wv[h][j] = sum_{l<16} qh[16h+l] * Wk[16h+l][j]
  float* wv = ws + WV_OFF + (size_t)b * 16 * 128;
  #pragma unroll
  for (int h = 0; h < 8; ++h) {
    float a = 0.f;
    #pragma unroll
    for (int l = 0; l < 16; ++l) a = fmaf(qh[16 * h + l], Wk[(size_t)(16 * h + l) * DK + tid], a);
    wv[h * 128 + tid] = a;
    wv[(8 + h) * 128 + tid] = 0.f;   // pad heads 8..15 for the 16-wide WMMA N dim
  }
  if (tid < 8) {
    float cc = 0.f;
    #pragma unroll
    for (int l = 0; l < 16; ++l) cc = fmaf(qh[16 * tid + l], bk[16 * tid + l], cc);
    ws[C_OFF + b * 8 + tid] = cc;
  }
}

// ---------------- K1: scores via WMMA f32 16x16x4 ----------------
// grid (B, T/64), 128 threads (4 waves); wave w computes t-rows [16w,16w+16)
__global__ void __launch_bounds__(128)
scores_kernel(const float* __restrict__ k, const int* __restrict__ start_ind,
              const int* __restrict__ end_ind, float* __restrict__ ws) {
  int b = blockIdx.x;
  int t0 = blockIdx.y * 64;
  int tid = threadIdx.x;
  __shared__ float kt[64 * 132];   // 64 t-rows x 128 feat, stride 132 (bank-conflict pad)
  __shared__ float wl[16 * 132];   // 16 "heads" x 128 feat

  // cooperative k tile load, b128 per thread
  const f4* kb4 = (const f4*)(k + ((size_t)b * T_ + t0) * DK);
  for (int i = tid; i < 64 * 32; i += 128) {
    int row = i >> 5, c4 = i & 31;
    *((f4*)(kt + row * 132) + c4) = kb4[(size_t)row * 32 + c4];
  }
  // prefetch next tile (speculative, dropped if OOB translation fails)
  if (t0 + 64 < T_)
    __builtin_prefetch((const char*)(kb4 + 64 * 32) + (size_t)tid * 256, 0, 1);
  const float* wv = ws + WV_OFF + (size_t)b * 16 * 128;
  for (int i = tid; i < 16 * 128; i += 128) wl[(i >> 7) * 132 + (i & 127)] = wv[i];
  __syncthreads();

  int wave = tid >> 5;
  int lane = tid & 31;
  int half = lane >> 4;       // K sub-block select per ISA A/B layout
  int lm   = lane & 15;       // A: M row; B: N col (head)
  int koff = half * 2;
  int trow = 16 * wave + lm;

  v8f acc = {};
  #pragma unroll 8
  for (int kk = 0; kk < DK; kk += 4) {
    v2f a, bb;
    a.x  = kt[trow * 132 + kk + koff];
    a.y  = kt[trow * 132 + kk + koff + 1];
    bb.x = wl[lm * 132 + kk + koff];
    bb.y = wl[lm * 132 + kk + koff + 1];
    acc = __builtin_amdgcn_wmma_f32_16x16x4_f32(false, a, false, bb, (short)0, acc,
                                                false, false);
  }

  // epilogue: C layout => lane holds head lm, rows M = r + 8*half
  if (lm < 8) {
    float ch = ws[C_OFF + b * 8 + lm];
    int s0 = start_ind[b], e0 = end_ind[b];
    int tbase = t0 + 16 * wave + 8 * half;
    float* srow = ws + S_OFF + ((size_t)b * 8 + lm) * T_ + tbase;
    #pragma unroll
    for (int r = 0; r < 8; ++r) {
      int t = tbase + r;
      float sc = (acc[r] + ch) * 0.25f;        // 1/sqrt(dk=16), temperature=1
      if (t < s0 || t > e0) sc = -3.0e38f;     // window mask -> exp==0
      srow[r] = sc;
    }
  }
}

// ---------------- K2: softmax stats per (b,h) ----------------
__global__ void __launch_bounds__(256)
stats_kernel(float* __restrict__ ws) {
  int bh = blockIdx.x, tid = threadIdx.x;
  const float* s = ws + S_OFF + (size_t)bh * T_;
  float vals[16];
  float m = -3.4e38f;
  #pragma unroll
  for (int i = 0; i < 16; ++i) { vals[i] = s[tid + 256 * i]; m = fmaxf(m, vals[i]); }
  #pragma unroll
  for (int off = 16; off > 0; off >>= 1) m = fmaxf(m, __shfl_xor(m, off, 32));
  __shared__ float red[8];
  if ((tid & 31) == 0) red[tid >> 5] = m;
  __syncthreads();
  float bm = red[0];
  #pragma unroll
  for (int i = 1; i < 8; ++i) bm = fmaxf(bm, red[i]);
  float sum = 0.f;
  #pragma unroll
  for (int i = 0; i < 16; ++i) sum += __expf(vals[i] - bm);
  #pragma unroll
  for (int off = 16; off > 0; off >>= 1) sum += __shfl_xor(sum, off, 32);
  __syncthreads();
  if ((tid & 31) == 0) red[tid >> 5] = sum;
  __syncthreads();
  if (tid == 0) {
    float tot = 0.f;
    #pragma unroll
    for (int i = 0; i < 8; ++i) tot += red[i];
    ws[RMAX_OFF + bh] = bm;
    ws[RSUM_OFF + bh] = 1.0f / tot;
  }
}

// ---------------- K3: attn weights + partial u ----------------
// grid (B, T/512), 256 threads
__global__ void __launch_bounds__(256)
attn_u_kernel(const float* __restrict__ v, float* __restrict__ amean,
              float* __restrict__ ws) {
  int b = blockIdx.x, chunk = blockIdx.y, tid = threadIdx.x;
  int t0 = chunk * 512;
  __shared__ float att[8][512];
  __shared__ float mx[8], sinv[8];
  if (tid < 8) {
    mx[tid]   = ws[RMAX_OFF + b * 8 + tid];
    sinv[tid] = ws[RSUM_OFF + b * 8 + tid];
  }
  __syncthreads();
  #pragma unroll
  for (int r = 0; r < 2; ++r) {
    int tl = r * 256 + tid;
    int t = t0 + tl;
    float msum = 0.f;
    #pragma unroll
    for (int h = 0; h < 8; ++h) {
      float sc = ws[S_OFF + ((size_t)b * 8 + h) * T_ + t];
      float a = __expf(sc - mx[h]) * sinv[h];
      att[h][tl] = a;
      msum += a;
    }
    amean[(size_t)b * T_ + t] = msum * 0.125f;
  }
  __syncthreads();
  // thread owns output column d = tid
  float acc[8] = {0.f, 0.f, 0.f, 0.f, 0.f, 0.f, 0.f, 0.f};
  const float* vb = v + ((size_t)b * T_ + t0) * D_ + tid;
  for (int tl = 0; tl < 512; ++tl) {
    float vv = vb[(size_t)tl * D_];
    #pragma unroll
    for (int h = 0; h < 8; ++h) acc[h] = fmaf(att[h][tl], vv, acc[h]);
  }
  float* up = ws + UPART_OFF + (((size_t)b * 8 + chunk) * 8) * 256 + tid;
  #pragma unroll
  for (int h = 0; h < 8; ++h) up[(size_t)h * 256] = acc[h];
}

// ---------------- K4: reduce partials + Wv/Wo projections ----------------
__global__ void __launch_bounds__(256)
outproj_kernel(const float* __restrict__ Wv, const float* __restrict__ bv,
               const float* __restrict__ Wo, const float* __restrict__ bo,
               const float* __restrict__ ws, float* __restrict__ out) {
  int b = blockIdx.x, o = threadIdx.x;
  __shared__ float u_l[8 * 256];
  __shared__ float vals[256];
  const float* up = ws + UPART_OFF + (size_t)b * 8 * 8 * 256;
  for (int i = o; i < 8 * 256; i += 256) {
    float s = 0.f;
    #pragma unroll
    for (int c = 0; c < 8; ++c) s += up[(size_t)c * 8 * 256 + i];  // fixed order
    u_l[i] = s;
  }
  __syncthreads();
  int h = o >> 5;                     // dv = 32 outputs per head
  const float* wr = Wv + (size_t)o * D_;
  const float* uh = u_l + h * 256;
  float a = bv[o];
  #pragma unroll 4
  for (int j = 0; j < D_; ++j) a = fmaf(wr[j], uh[j], a);
  vals[o] = a;
  __syncthreads();
  const float* wo = Wo + (size_t)o * D_;
  float r = bo[o];
  #pragma unroll 4
  for (int j = 0; j < D_; ++j) r = fmaf(wo[j], vals[j], r);
  out[(size_t)b * D_ + o] = r;
}

extern "C" void kernel_launch(void* const* d_in, const int* in_sizes, int n_in,
                              void* d_out, int out_size, void* d_ws, size_t ws_size,
                              hipStream_t stream) {
  const float* q  = (const float*)d_in[0];
  const float* k  = (const float*)d_in[1];
  const float* v  = (const float*)d_in[2];
  const float* Wq = (const float*)d_in[3];
  const float* bq = (const float*)d_in[4];
  const float* Wk = (const float*)d_in[5];
  const float* bk = (const float*)d_in[6];
  const float* Wv = (const float*)d_in[7];
  const float* bv = (const float*)d_in[8];
  const float* Wo = (const float*)d_in[9];
  const float* bo = (const float*)d_in[10];
  const int* si   = (const int*)d_in[11];
  const int* ei   = (const int*)d_in[12];
  float* ws   = (float*)d_ws;
  float* out  = (float*)d_out;             // [B, D]
  float* am   = out + (size_t)B_ * D_;     // [B, T] head-averaged attention

  prep_kernel  <<<B_,                 128, 0, stream>>>(q, Wq, bq, Wk, bk, ws);
  scores_kernel<<<dim3(B_, T_ / 64),  128, 0, stream>>>(k, si, ei, ws);
  stats_kernel <<<B_ * H_,            256, 0, stream>>>(ws);
  attn_u_kernel<<<dim3(B_, T_ / 512), 256, 0, stream>>>(v, am, ws);
  outproj_kernel<<<B_,                256, 0, stream>>>(Wv, bv, Wo, bo, ws, out);
}